// SimAttention_47923245089133
// MI455X (gfx1250) — compile-verified
//
#include <hip/hip_runtime.h>

typedef float v2f __attribute__((ext_vector_type(2)));
typedef float v8f __attribute__((ext_vector_type(8)));

#define Bb 16
#define Tt 2048
#define Jj 512
#define Dd 256
#define BM 128
#define BN 128
#define BK 32
#define LDK 36   // padded LDS row stride in floats (16B aligned, bank-conflict-free)

// CDNA5 async global->LDS copy, 16 bytes per lane (GVS mode):
//   global address = 64-bit SGPR base + 32-bit unsigned VGPR offset + imm offset
//   data is written to LDS at the VDST address + imm offset  (tracked by ASYNCcnt)
__device__ __forceinline__ void async_ld16(unsigned lds_addr, const float* sbase,
                                           unsigned voff) {
  asm volatile("global_load_async_to_lds_b128 %0, %1, %2 offset:0"
               :: "v"(lds_addr), "v"(voff), "s"(sbase) : "memory");
}
__device__ __forceinline__ void wait_async0() {
  asm volatile("s_wait_asynccnt 0" ::: "memory");
}

__global__ __launch_bounds__(256) void
simattn_wmma_f32(const float* __restrict__ H,
                 const float* __restrict__ U,
                 const float* __restrict__ W,
                 float* __restrict__ out) {
  __shared__ float Hs[2][BM * LDK];
  __shared__ float Us[2][BN * LDK];
  __shared__ float ws[3 * Dd];   // w_h | w_u | w_m
  __shared__ float sh[BM];
  __shared__ float su[BN];

  const int tid  = threadIdx.x;
  const int lane = tid & 31;
  const int wid  = tid >> 5;
  const int b    = blockIdx.y;
  const int tileM = blockIdx.x % (Tt / BM);   // 0..15
  const int tileN = blockIdx.x / (Tt / BM);   // 0..3
  const int t0 = tileM * BM;
  const int j0 = tileN * BN;

  for (int i = tid; i < 3 * Dd; i += 256) ws[i] = W[i];
  if (tid < BM) sh[tid] = 0.0f; else su[tid - BM] = 0.0f;

  const float* Hb = H + (size_t)b * Tt * Dd + (size_t)t0 * Dd;
  const float* Ub = U + (size_t)b * Jj * Dd + (size_t)j0 * Dd;

  // per-thread staging geometry: 1024 float4 per 128x32 slab, 8 float4 per row
  unsigned gvoff[4];    // global byte offset of this thread's float4 (per r)
  unsigned lsoff[4];    // LDS byte offset within a tile buffer (per r)
#pragma unroll
  for (int r = 0; r < 4; ++r) {
    const int f4  = tid + 256 * r;
    const int row = f4 >> 3;
    const int c4  = f4 & 7;
    gvoff[r] = (unsigned)((row * Dd + c4 * 4) * 4);
    lsoff[r] = (unsigned)((row * LDK + c4 * 4) * 4);
  }
  const unsigned hsb[2] = { (unsigned)(uintptr_t)&Hs[0][0],
                            (unsigned)(uintptr_t)&Hs[1][0] };
  const unsigned usb[2] = { (unsigned)(uintptr_t)&Us[0][0],
                            (unsigned)(uintptr_t)&Us[1][0] };

  // wave tiling: 4 waves along M (32 rows each), 2 waves along N (64 cols each)
  const int wm0 = (wid & 3) * 32;
  const int wn0 = (wid >> 2) * 64;
  const int hi  = lane >> 4;      // 0: K pair {0,1}; 1: K pair {2,3}
  const int l16 = lane & 15;

  v8f acc[2][4];
  const v8f vzero = {0.f, 0.f, 0.f, 0.f, 0.f, 0.f, 0.f, 0.f};
#pragma unroll
  for (int mt = 0; mt < 2; ++mt)
#pragma unroll
    for (int nt = 0; nt < 4; ++nt) acc[mt][nt] = vzero;

  // ---- prologue: async-stage slab 0 into buffer 0 ----
#pragma unroll
  for (int r = 0; r < 4; ++r) {
    async_ld16(hsb[0] + lsoff[r], Hb, gvoff[r]);
    async_ld16(usb[0] + lsoff[r], Ub, gvoff[r]);
  }
  wait_async0();
  __syncthreads();

  for (int kb = 0; kb < Dd / BK; ++kb) {
    const int cur = kb & 1;
    const int k0  = kb * BK;

    // ---- async-stage next slab into the other buffer (overlaps compute) ----
    if (kb + 1 < Dd / BK) {
      const int nxt = cur ^ 1;
      const float* Hn = Hb + (kb + 1) * BK;
      const float* Un = Ub + (kb + 1) * BK;
#pragma unroll
      for (int r = 0; r < 4; ++r) {
        async_ld16(hsb[nxt] + lsoff[r], Hn, gvoff[r]);
        async_ld16(usb[nxt] + lsoff[r], Un, gvoff[r]);
      }
    }

    // ---- rank-1 bias partials from the same LDS tiles (wave-uniform split) ----
    if (tid < BM) {
      float a = 0.0f;
#pragma unroll
      for (int kk = 0; kk < BK; ++kk) a += Hs[cur][tid * LDK + kk] * ws[k0 + kk];
      sh[tid] += a;
    } else {
      const int i = tid - BM;
      float a = 0.0f;
#pragma unroll
      for (int kk = 0; kk < BK; ++kk) a += Us[cur][i * LDK + kk] * ws[Dd + k0 + kk];
      su[i] += a;
    }

    // ---- WMMA over this K slab: 8 steps of K=4, 8 wmma each ----
#pragma unroll
    for (int kk = 0; kk < BK; kk += 4) {
      const int kb4 = kk + hi * 2;                 // lane's K pair within slab
      v2f wmv;
      wmv.x = ws[2 * Dd + k0 + kb4];
      wmv.y = ws[2 * Dd + k0 + kb4 + 1];

      v2f afrag[2], bfrag[4];
#pragma unroll
      for (int mt = 0; mt < 2; ++mt) {
        v2f t = *(const v2f*)(&Hs[cur][(wm0 + mt * 16 + l16) * LDK + kb4]);
        afrag[mt].x = t.x * wmv.x;                 // fold w_m into A
        afrag[mt].y = t.y * wmv.y;
      }
#pragma unroll
      for (int nt = 0; nt < 4; ++nt)
        bfrag[nt] = *(const v2f*)(&Us[cur][(wn0 + nt * 16 + l16) * LDK + kb4]);

#pragma unroll
      for (int mt = 0; mt < 2; ++mt)
#pragma unroll
        for (int nt = 0; nt < 4; ++nt)
          acc[mt][nt] = __builtin_amdgcn_wmma_f32_16x16x4_f32(
              /*neg_a=*/false, afrag[mt], /*neg_b=*/false, bfrag[nt],
              /*c_mod=*/(short)0, acc[mt][nt],
              /*reuse_a=*/false, /*reuse_b=*/false);
    }

    // own async writes done, then make them visible to all waves
    wait_async0();
    __syncthreads();
  }

  // ---- epilogue: out = acc + sh[row] + su[col] ----
  float suv[4];
#pragma unroll
  for (int nt = 0; nt < 4; ++nt) suv[nt] = su[wn0 + nt * 16 + l16];

  float* outp = out + (size_t)b * Tt * Jj;
#pragma unroll
  for (int mt = 0; mt < 2; ++mt) {
#pragma unroll
    for (int r = 0; r < 8; ++r) {
      const int mrow = wm0 + mt * 16 + hi * 8 + r;
      const float shv = sh[mrow];
      const size_t rowoff = (size_t)(t0 + mrow) * Jj;
#pragma unroll
      for (int nt = 0; nt < 4; ++nt) {
        const int col = j0 + wn0 + nt * 16 + l16;
        outp[rowoff + col] = acc[mt][nt][r] + shv + suv[nt];
      }
    }
  }
}

extern "C" void kernel_launch(void* const* d_in, const int* in_sizes, int n_in,
                              void* d_out, int out_size, void* d_ws, size_t ws_size,
                              hipStream_t stream) {
  const float* H = (const float*)d_in[0];   // (B, T, D) f32
  const float* U = (const float*)d_in[1];   // (B, J, D) f32
  const float* W = (const float*)d_in[2];   // (3D, 1)  f32
  float* out = (float*)d_out;               // (B, T, J) f32

  dim3 grid((Tt / BM) * (Jj / BN), Bb);     // (64, 16)
  simattn_wmma_f32<<<grid, 256, 0, stream>>>(H, U, W, out);
}